// DTIHarmonic_32770600468663
// MI455X (gfx1250) — compile-verified
//
#include <hip/hip_runtime.h>
#include <math.h>

#define DIM   128
#define NB    8
#define NN1   128
#define NN2   512
#define NGNN  3
#define KPAD  64   // node-feature K (54) padded to 64

typedef float v2f __attribute__((ext_vector_type(2)));
typedef float v8f __attribute__((ext_vector_type(8)));

// ---------------------------------------------------------------------------
// Zero-padding copy: dst[dstRows,dstCols] = src[srcRows,srcCols] zero-extended
// ---------------------------------------------------------------------------
__global__ __launch_bounds__(256) void pad2d_kernel(
    const float* __restrict__ src, float* __restrict__ dst,
    int srcRows, int srcCols, int dstRows, int dstCols)
{
  const int idx = blockIdx.x * 256 + threadIdx.x;
  const int total = dstRows * dstCols;
  if (idx >= total) return;
  const int r = idx / dstCols;
  const int c = idx - r * dstCols;
  dst[idx] = (r < srcRows && c < srcCols) ? src[(long)r * srcCols + c] : 0.f;
}

// ---------------------------------------------------------------------------
// Batched GEMM on the fp32 WMMA pipe: C = A * op(B) (+bias)(+relu).
// One wave per 16x16 C tile, V_WMMA_F32_16X16X4_F32, K multiple of 4,
// unconditional b64 fragment loads, K-loop unrolled x4.
// Fragment layouts per CDNA5 ISA 7.12.2:
//   A (16x4):  lanes0-15 rows M with K{0,1}; lanes16-31 K{2,3}
//   B (4x16):  vgpr0: K=0(lo)/K=2(hi), vgpr1: K=1(lo)/K=3(hi), col = lane&15
//   C (16x16): vgpr r -> row m0+r+8*hi, col n0+(lane&15)
// ---------------------------------------------------------------------------
template <int TRANSB>
__global__ __launch_bounds__(256) void wmma_gemm_f32(
    const float* __restrict__ A, int lda, long sA,
    const float* __restrict__ Bm, int ldb, long sB,
    float* __restrict__ C, int ldc, long sC,
    const float* __restrict__ bias, int relu,
    int M, int N, int K)
{
  __builtin_assume(K >= 4 && (K & 3) == 0);
  const int b    = blockIdx.z;
  const int wave = threadIdx.x >> 5;
  const int lane = threadIdx.x & 31;
  const int l15  = lane & 15;
  const int hi   = lane >> 4;           // 0: lanes 0-15, 1: lanes 16-31

  const int m0 = blockIdx.x * 16;
  const int n0 = blockIdx.y * 128 + wave * 16;
  if (m0 >= M || n0 >= N) return;       // wave-uniform guard (EXEC stays full)

  const int rowA = m0 + l15;
  const int coln = n0 + l15;

  const float* __restrict__ Arow = A + (long)b * sA + (long)rowA * lda + 2 * hi;
  const float* __restrict__ Bp;
  if (TRANSB) {                          // logical B[k][n] = Bt[n][k]
    Bp = Bm + (long)b * sB + (long)coln * ldb + 2 * hi;
  } else {
    Bp = Bm + (long)b * sB + (long)(2 * hi) * ldb + coln;
  }

  v8f acc = {0.f, 0.f, 0.f, 0.f, 0.f, 0.f, 0.f, 0.f};

#pragma unroll 4
  for (int k0 = 0; k0 < K; k0 += 4) {
    const v2f afr = *(const v2f*)(Arow + k0);
    v2f bfr;
    if (TRANSB) {
      bfr = *(const v2f*)(Bp + k0);
    } else {
      const float* bq = Bp + (long)k0 * ldb;
      bfr.x = bq[0];
      bfr.y = bq[ldb];
    }
    acc = __builtin_amdgcn_wmma_f32_16x16x4_f32(
        false, afr, false, bfr, (short)0, acc, false, false);
  }

  float* __restrict__ Cb = C + (long)b * sC;
  const float bv = bias ? bias[coln] : 0.f;
#pragma unroll
  for (int r = 0; r < 8; ++r) {
    float v = acc[r] + bv;
    if (relu) v = fmaxf(v, 0.f);
    Cb[(long)(m0 + r + 8 * hi) * ldc + coln] = v;
  }
}

// ---------------------------------------------------------------------------
// Masked column softmax of (e + e^T), axis=1 (over rows i for fixed col k),
// multiplied elementwise by adj. One block per (k, b).
// ---------------------------------------------------------------------------
__global__ __launch_bounds__(256) void gat_att_softmax(
    const float* __restrict__ e, const float* __restrict__ adj,
    float* __restrict__ att, int N)
{
  const int k = blockIdx.x;
  const int b = blockIdx.y;
  const float* eb   = e   + (long)b * N * N;
  const float* adjb = adj + (long)b * N * N;
  float*       atb  = att + (long)b * N * N;

  __shared__ float red[256];
  const int t = threadIdx.x;

  float mmax = -3.0e38f;
  for (int i = t; i < N; i += 256) {
    const float a = adjb[(long)i * N + k];
    const float v = (a > 0.f) ? (eb[(long)i * N + k] + eb[(long)k * N + i])
                              : -9.0e15f;
    mmax = fmaxf(mmax, v);
  }
  red[t] = mmax; __syncthreads();
  for (int s = 128; s > 0; s >>= 1) {
    if (t < s) red[t] = fmaxf(red[t], red[t + s]);
    __syncthreads();
  }
  mmax = red[0]; __syncthreads();

  float ssum = 0.f;
  for (int i = t; i < N; i += 256) {
    const float a = adjb[(long)i * N + k];
    const float v = (a > 0.f) ? (eb[(long)i * N + k] + eb[(long)k * N + i])
                              : -9.0e15f;
    ssum += expf(v - mmax);
  }
  red[t] = ssum; __syncthreads();
  for (int s = 128; s > 0; s >>= 1) {
    if (t < s) red[t] += red[t + s];
    __syncthreads();
  }
  const float inv = 1.f / red[0];

  for (int i = t; i < N; i += 256) {
    const float a = adjb[(long)i * N + k];
    const float v = (a > 0.f) ? (eb[(long)i * N + k] + eb[(long)k * N + i])
                              : -9.0e15f;
    atb[(long)i * N + k] = expf(v - mmax) * inv * a;
  }
}

// ---------------------------------------------------------------------------
// Gated residual: coeff = sigmoid([x,h']·gW + gb); out = c*x + (1-c)*h'.
// One wave32 per node; cross-lane reduce with __shfl_xor.
// ---------------------------------------------------------------------------
__global__ __launch_bounds__(256) void gat_gate(
    const float* __restrict__ x, const float* __restrict__ hp,
    const float* __restrict__ gW, const float* __restrict__ gB, int layer,
    float* __restrict__ out, int totalNodes)
{
  const int wave = threadIdx.x >> 5;
  const int lane = threadIdx.x & 31;
  const int node = blockIdx.x * 8 + wave;
  if (node >= totalNodes) return;

  const float* xr = x  + (long)node * DIM;
  const float* hr = hp + (long)node * DIM;

  float s = 0.f;
#pragma unroll
  for (int tt = 0; tt < 8; ++tt) {
    const int d = lane + 32 * tt;
    const float v = (d < DIM) ? xr[d] : hr[d - DIM];
    s += v * gW[d];
  }
#pragma unroll
  for (int off = 16; off > 0; off >>= 1) s += __shfl_xor(s, off, 32);

  const float coeff = 1.f / (1.f + expf(-(s + gB[layer])));
  float* orow = out + (long)node * DIM;
#pragma unroll
  for (int tt = 0; tt < 4; ++tt) {
    const int d = lane + 32 * tt;
    orow[d] = coeff * xr[d] + (1.f - coeff) * hr[d];
  }
}

// ---------------------------------------------------------------------------
// Fused pair kernel: per (b,i) block, iterate j; ReLU-MLP dot over 128 dims
// (A and B heads, float4 b128 loads), distance terms, vdW/hbond/hydrophobic
// partial sums -> LDS tree -> atomicAdd.
// ---------------------------------------------------------------------------
__global__ __launch_bounds__(256) void pair_energy(
    const float* __restrict__ p1A, const float* __restrict__ p1B,
    const float* __restrict__ p2A, const float* __restrict__ p2B,
    const float* __restrict__ w2A, const float* __restrict__ b2A,
    const float* __restrict__ w2B, const float* __restrict__ b2B,
    const float* __restrict__ pos1, const float* __restrict__ pos2,
    const float* __restrict__ A_int,
    const float* __restrict__ r1, const float* __restrict__ r2,
    const float* __restrict__ nm1, const float* __restrict__ nm2,
    float* __restrict__ acc)
{
  const int i = blockIdx.x;      // 0..127
  const int b = blockIdx.y;      // 0..7
  const int t = threadIdx.x;

  __shared__ float s1A[DIM], s1B[DIM], swA[DIM], swB[DIM];
  __shared__ float red[4][256];

  const long n1idx = (long)b * NN1 + i;
  if (t < DIM) {
    s1A[t] = p1A[n1idx * DIM + t];
    s1B[t] = p1B[n1idx * DIM + t];
  } else {
    swA[t - DIM] = w2A[t - DIM];
    swB[t - DIM] = w2B[t - DIM];
  }
  __syncthreads();

  const float px = pos1[n1idx * 3 + 0];
  const float py = pos1[n1idx * 3 + 1];
  const float pz = pos1[n1idx * 3 + 2];
  const float rad1 = r1[n1idx];
  const float m1   = nm1[n1idx];
  const float bbA  = b2A[0];
  const float bbB  = b2B[0];

  float a_vdw = 0.f, a_hb1 = 0.f, a_hb2 = 0.f, a_hp = 0.f;

  for (int j = t; j < NN2; j += 256) {
    const long n2idx = (long)b * NN2 + j;
    const float4* q2A = (const float4*)(p2A + n2idx * DIM);
    const float4* q2B = (const float4*)(p2B + n2idx * DIM);

    float zA = 0.f, zB = 0.f;
#pragma unroll 4
    for (int h4 = 0; h4 < DIM / 4; ++h4) {
      const float4 a4 = q2A[h4];
      const float4 b4 = q2B[h4];
      const int h = 4 * h4;
      zA += fmaxf(s1A[h + 0] + a4.x, 0.f) * swA[h + 0];
      zA += fmaxf(s1A[h + 1] + a4.y, 0.f) * swA[h + 1];
      zA += fmaxf(s1A[h + 2] + a4.z, 0.f) * swA[h + 2];
      zA += fmaxf(s1A[h + 3] + a4.w, 0.f) * swA[h + 3];
      zB += fmaxf(s1B[h + 0] + b4.x, 0.f) * swB[h + 0];
      zB += fmaxf(s1B[h + 1] + b4.y, 0.f) * swB[h + 1];
      zB += fmaxf(s1B[h + 2] + b4.z, 0.f) * swB[h + 2];
      zB += fmaxf(s1B[h + 3] + b4.w, 0.f) * swB[h + 3];
    }

    const float dx = px - pos2[n2idx * 3 + 0];
    const float dy = py - pos2[n2idx * 3 + 1];
    const float dz = pz - pos2[n2idx * 3 + 2];
    float dm = sqrtf(dx * dx + dy * dy + dz * dz + 1e-10f);
    if (dm < 0.5f) dm = 1.0e10f;

    const float Aw = 1.f / (1.f + expf(-(zA + bbA)));
    const float Bw = tanhf(zB + bbB) * 0.2f;

    const float dm0  = rad1 + r2[n2idx] + Bw;
    const float dm0s = (dm0 < 1e-4f) ? 1.f : dm0;
    const float rr   = dm0s / dm;
    const float rr2  = rr * rr;
    const float rN   = rr2 * rr2 * rr2;                 // (dm0/dm)^6
    const float evdw = fminf(rN * rN - 2.f * rN, 100.f);
    const float Aamp = Aw * (0.0356f - 0.0178f) + 0.0178f;
    a_vdw += Aamp * evdw * (m1 * nm2[n2idx]);

    const float dmd  = dm - dm0;
    const long abase = (((long)b * 8) * NN1 + i) * NN2 + j;
    const float A1 = A_int[abase + (long)1 * NN1 * NN2];
    const float A7 = A_int[abase + (long)7 * NN1 * NN2];
    const float A6 = A_int[abase + (long)6 * NN1 * NN2];
    a_hb1 += fminf(fmaxf(dmd * A1 * (-1.f / 0.7f), 0.f), 1.f);
    a_hb2 += fminf(fmaxf(dmd * A7 * (-1.f / 0.7f), 0.f), 1.f);
    a_hp  += fminf(fmaxf((-dmd + 1.5f) * A6, 0.f), 1.f);
  }

  red[0][t] = a_vdw; red[1][t] = a_hb1; red[2][t] = a_hb2; red[3][t] = a_hp;
  __syncthreads();
  for (int s = 128; s > 0; s >>= 1) {
    if (t < s) {
      red[0][t] += red[0][t + s];
      red[1][t] += red[1][t + s];
      red[2][t] += red[2][t + s];
      red[3][t] += red[3][t + s];
    }
    __syncthreads();
  }
  if (t == 0) {
    atomicAdd(&acc[b * 4 + 0], red[0][0]);
    atomicAdd(&acc[b * 4 + 1], red[1][0]);
    atomicAdd(&acc[b * 4 + 2], red[2][0]);
    atomicAdd(&acc[b * 4 + 3], red[3][0]);
  }
}

__global__ void zero_acc_kernel(float* acc) {
  if (threadIdx.x < 64) acc[threadIdx.x] = 0.f;
}

__global__ void finalize_kernel(
    const float* __restrict__ acc, const float* __restrict__ rotor,
    const float* __restrict__ delta_uff, const float* __restrict__ hbond,
    const float* __restrict__ hydro, const float* __restrict__ vdwc,
    const float* __restrict__ rotc, float* __restrict__ out)
{
  const int b = threadIdx.x;
  if (b >= NB) return;
  const float hb = -(hbond[0] * hbond[0]);
  const float hy = -(hydro[0] * hydro[0]);
  const float denom = 1.f + rotc[0] * rotc[0] * rotor[b];
  out[b * 5 + 0] = acc[b * 4 + 0] / denom;
  out[b * 5 + 1] = acc[b * 4 + 1] * hb / denom;
  out[b * 5 + 2] = acc[b * 4 + 2] * hb / denom;
  out[b * 5 + 3] = acc[b * 4 + 3] * hy / denom;
  out[b * 5 + 4] = delta_uff[b] * (vdwc[0] * vdwc[0]) / denom;
}

// ---------------------------------------------------------------------------
// Host side
// ---------------------------------------------------------------------------
static inline void gemm(hipStream_t st,
                        const float* A, int lda, long sA,
                        const float* B, int ldb, long sB, int tB,
                        float* C, int ldc, long sC,
                        const float* bias, int relu,
                        int M, int N, int K, int batch)
{
  dim3 g(M / 16, (N + 127) / 128, batch);
  if (tB)
    wmma_gemm_f32<1><<<g, dim3(256), 0, st>>>(A, lda, sA, B, ldb, sB,
                                              C, ldc, sC, bias, relu, M, N, K);
  else
    wmma_gemm_f32<0><<<g, dim3(256), 0, st>>>(A, lda, sA, B, ldb, sB,
                                              C, ldc, sC, bias, relu, M, N, K);
}

extern "C" void kernel_launch(void* const* d_in, const int* in_sizes, int n_in,
                              void* d_out, int out_size, void* d_ws, size_t ws_size,
                              hipStream_t stream) {
  const float* h1     = (const float*)d_in[0];
  const float* adj1   = (const float*)d_in[1];
  const float* h2     = (const float*)d_in[2];
  const float* adj2   = (const float*)d_in[3];
  const float* A_int  = (const float*)d_in[4];
  const float* pos1   = (const float*)d_in[5];
  const float* pos2   = (const float*)d_in[6];
  const float* rotor  = (const float*)d_in[7];
  const float* vdwr1  = (const float*)d_in[8];
  const float* vdwr2  = (const float*)d_in[9];
  const float* d_uff  = (const float*)d_in[10];
  const float* nm1    = (const float*)d_in[11];
  const float* nm2    = (const float*)d_in[12];
  const float* node_W = (const float*)d_in[13];
  const float* gat_W  = (const float*)d_in[14];
  const float* gat_b  = (const float*)d_in[15];
  const float* gat_A  = (const float*)d_in[16];
  const float* gate_W = (const float*)d_in[17];
  const float* gate_b = (const float*)d_in[18];
  const float* vA_W1  = (const float*)d_in[19];
  const float* vA_b1  = (const float*)d_in[20];
  const float* vA_W2  = (const float*)d_in[21];
  const float* vA_b2  = (const float*)d_in[22];
  const float* vB_W1  = (const float*)d_in[23];
  const float* vB_b1  = (const float*)d_in[24];
  const float* vB_W2  = (const float*)d_in[25];
  const float* vB_b2  = (const float*)d_in[26];
  const float* hbond  = (const float*)d_in[27];
  const float* hydro  = (const float*)d_in[28];
  const float* vdwc   = (const float*)d_in[29];
  const float* rotc   = (const float*)d_in[30];
  float* out = (float*)d_out;

  // workspace layout (floats)
  float* ws = (float*)d_ws;
  const long S1  = (long)NB * NN1 * DIM;   // 131072
  const long S2  = (long)NB * NN2 * DIM;   // 524288
  const long E1S = (long)NB * NN1 * NN1;   // 131072
  const long E2S = (long)NB * NN2 * NN2;   // 2097152
  long o = 0;
  float* X1A = ws + o; o += S1;
  float* X1B = ws + o; o += S1;
  float* X2A = ws + o; o += S2;
  float* X2B = ws + o; o += S2;
  float* H1  = ws + o; o += S1;
  float* HA1 = ws + o; o += S1;
  float* H2  = ws + o; o += S2;
  float* HA2 = ws + o; o += S2;
  float* E1  = ws + o; o += E1S;
  float* AT1 = ws + o; o += E1S;
  float* E2  = ws + o; o += E2S;
  float* AT2 = ws + o; o += E2S;
  float* PH1 = ws + o; o += (long)NB * NN1 * KPAD;   // padded h1
  float* PH2 = ws + o; o += (long)NB * NN2 * KPAD;   // padded h2
  float* PNW = ws + o; o += (long)KPAD * DIM;        // padded node_W
  float* ACC = ws + o; o += 64;
  (void)ws_size; (void)in_sizes; (void)n_in; (void)out_size;

  // pad node features K=54 -> 64 and node_W rows 54 -> 64 (zero fill)
  {
    const int t1 = NB * NN1 * KPAD;
    pad2d_kernel<<<(t1 + 255) / 256, 256, 0, stream>>>(h1, PH1, NB * NN1, 54,
                                                       NB * NN1, KPAD);
    const int t2 = NB * NN2 * KPAD;
    pad2d_kernel<<<(t2 + 255) / 256, 256, 0, stream>>>(h2, PH2, NB * NN2, 54,
                                                       NB * NN2, KPAD);
    const int t3 = KPAD * DIM;
    pad2d_kernel<<<(t3 + 255) / 256, 256, 0, stream>>>(node_W, PNW, 54, DIM,
                                                       KPAD, DIM);
  }

  // node embedding: g = h_pad @ node_W_pad  (K=64, unguarded)
  gemm(stream, PH1, KPAD, (long)NN1 * KPAD, PNW, DIM, 0, 0,
       X1A, DIM, (long)NN1 * DIM, nullptr, 0, NN1, DIM, KPAD, NB);
  gemm(stream, PH2, KPAD, (long)NN2 * KPAD, PNW, DIM, 0, 0,
       X2A, DIM, (long)NN2 * DIM, nullptr, 0, NN2, DIM, KPAD, NB);

  float* x1 = X1A; float* x1n = X1B;
  float* x2 = X2A; float* x2n = X2B;
  const long sA1 = (long)NN1 * DIM, sA2 = (long)NN2 * DIM;

  for (int l = 0; l < NGNN; ++l) {
    const float* W  = gat_W  + (long)l * DIM * DIM;
    const float* bv = gat_b  + (long)l * DIM;
    const float* Am = gat_A  + (long)l * DIM * DIM;
    const float* gW = gate_W + (long)l * 2 * DIM;

    // ---- graph 1 (N=128) ----
    gemm(stream, x1, DIM, sA1, W, DIM, 0, 0, H1, DIM, sA1, bv, 0, NN1, DIM, DIM, NB);
    gemm(stream, H1, DIM, sA1, Am, DIM, 0, 0, HA1, DIM, sA1, nullptr, 0, NN1, DIM, DIM, NB);
    gemm(stream, HA1, DIM, sA1, H1, DIM, sA1, 1, E1, NN1, E1S / NB, nullptr, 0, NN1, NN1, DIM, NB);
    gat_att_softmax<<<dim3(NN1, NB), 256, 0, stream>>>(E1, adj1, AT1, NN1);
    gemm(stream, AT1, NN1, E1S / NB, H1, DIM, sA1, 0, HA1, DIM, sA1, nullptr, 1, NN1, DIM, NN1, NB);
    gat_gate<<<dim3(NB * NN1 / 8), 256, 0, stream>>>(x1, HA1, gW, gate_b, l, x1n, NB * NN1);

    // ---- graph 2 (N=512) ----
    gemm(stream, x2, DIM, sA2, W, DIM, 0, 0, H2, DIM, sA2, bv, 0, NN2, DIM, DIM, NB);
    gemm(stream, H2, DIM, sA2, Am, DIM, 0, 0, HA2, DIM, sA2, nullptr, 0, NN2, DIM, DIM, NB);
    gemm(stream, HA2, DIM, sA2, H2, DIM, sA2, 1, E2, NN2, E2S / NB, nullptr, 0, NN2, NN2, DIM, NB);
    gat_att_softmax<<<dim3(NN2, NB), 256, 0, stream>>>(E2, adj2, AT2, NN2);
    gemm(stream, AT2, NN2, E2S / NB, H2, DIM, sA2, 0, HA2, DIM, sA2, nullptr, 1, NN2, DIM, NN2, NB);
    gat_gate<<<dim3(NB * NN2 / 8), 256, 0, stream>>>(x2, HA2, gW, gate_b, l, x2n, NB * NN2);

    float* tmp;
    tmp = x1; x1 = x1n; x1n = tmp;
    tmp = x2; x2 = x2n; x2n = tmp;
  }

  // pair projections (reuse H1/HA1/H2/HA2 as P1A/P1B/P2A/P2B)
  gemm(stream, x1, DIM, sA1, vA_W1, DIM, 0, 0, H1,  DIM, sA1, vA_b1,   0, NN1, DIM, DIM, NB);
  gemm(stream, x1, DIM, sA1, vB_W1, DIM, 0, 0, HA1, DIM, sA1, vB_b1,   0, NN1, DIM, DIM, NB);
  gemm(stream, x2, DIM, sA2, vA_W1 + DIM * DIM, DIM, 0, 0, H2,  DIM, sA2, nullptr, 0, NN2, DIM, DIM, NB);
  gemm(stream, x2, DIM, sA2, vB_W1 + DIM * DIM, DIM, 0, 0, HA2, DIM, sA2, nullptr, 0, NN2, DIM, DIM, NB);

  zero_acc_kernel<<<1, 64, 0, stream>>>(ACC);
  pair_energy<<<dim3(NN1, NB), 256, 0, stream>>>(
      H1, HA1, H2, HA2, vA_W2, vA_b2, vB_W2, vB_b2,
      pos1, pos2, A_int, vdwr1, vdwr2, nm1, nm2, ACC);
  finalize_kernel<<<1, NB, 0, stream>>>(ACC, rotor, d_uff, hbond, hydro, vdwc, rotc, out);
}